// GeoFormer_47373489274897
// MI455X (gfx1250) — compile-verified
//
#include <hip/hip_runtime.h>
#include <hip/hip_bf16.h>

// GeoFormer.generate_proposal for MI455X (gfx1250).
// Bandwidth-bound: ~226 MB of HBM traffic -> ~9.7us floor at 23.3 TB/s.
// gfx1250-specific paths: global_load_async_to_lds_b128 th:TH_LOAD_NT
// (ASYNCcnt DMA), s_wait_asynccnt split counters, global_prefetch_b8,
// NT store hints, wave32 shuffles.

typedef float v4f __attribute__((ext_vector_type(4)));
typedef int   v4i __attribute__((ext_vector_type(4)));

#define B_    4
#define Q_    128
#define N_    50000
#define C_    20
#define ROWS  (B_ * Q_)          // 512
#define N4    (N_ / 4)           // 12500 float4 per row
#define SPLIT 4                  // parts per row
#define PART4 (N4 / SPLIT)       // 3125 float4 per part
#define TPB   256
#define NITER ((PART4 + TPB - 1) / TPB)  // 13
#define DEPTH 8                  // async pipeline depth (per-wave buffers)
#define NWAVES (TPB / 32)

#define THRESH_CNT 50
#define MIN_CLS    4

// ---- CDNA5 async global->LDS DMA (tracked by ASYNCcnt) ------------------
// Read-once stream: non-temporal so it does not evict the hot seg_pred
// working set (800 KB re-read 128x) from L2.
__device__ __forceinline__ void async_b128(unsigned lds_off, const void* g) {
  asm volatile("global_load_async_to_lds_b128 %0, %1, off th:TH_LOAD_NT"
               :: "v"(lds_off), "v"((unsigned long long)g)
               : "memory");
}
template <int Nw> __device__ __forceinline__ void wait_async() {
  asm volatile("s_wait_asynccnt %0" :: "n"(Nw) : "memory");
}
// Generic pointer to a __shared__ object is {aperture_hi32, lds_offset_lo32};
// the async op wants the wave-relative LDS byte offset = low 32 bits.
__device__ __forceinline__ unsigned lds_off_of(const void* p) {
  return (unsigned)(unsigned long long)p;
}

// ---- Kernel A: argmax(cls_logits) + global_ids copy ---------------------
__global__ __launch_bounds__(TPB) void gf_prep(
    const float* __restrict__ cls_logits, const int* __restrict__ fg_idxs,
    int* __restrict__ ws_cls, float* __restrict__ cls_out,
    float* __restrict__ gids_out) {
  int t = blockIdx.x * TPB + threadIdx.x;
  if (t < B_ * N_) gids_out[t] = (float)fg_idxs[t];   // ints < 2^24: exact
  if (t < ROWS) {
    const float* cl = cls_logits + t * C_;
    float best = cl[0];
    int bi = 0;
#pragma unroll
    for (int c = 1; c < C_; ++c) {
      float v = cl[c];
      if (v > best) { best = v; bi = c; }  // first-max, matches jnp.argmax
    }
    ws_cls[t] = bi;
    cls_out[t] = (float)bi;
  }
}

// ---- Kernel B: streaming pass: write sel, reduce cnt/sum per part -------
__global__ __launch_bounds__(TPB) void gf_main(
    const float* __restrict__ mask_logits, const int* __restrict__ seg_pred,
    const int* __restrict__ ws_cls, float* __restrict__ masks_out,
    int* __restrict__ ws_pcnt, float* __restrict__ ws_psum) {
  __shared__ v4f stage[DEPTH][NWAVES][32];  // 32 KB, per-wave private bufs
  __shared__ int s_cnt[NWAVES];
  __shared__ float s_sum[NWAVES];

  const int row  = blockIdx.x / SPLIT;
  const int part = blockIdx.x % SPLIT;
  const int b    = row / Q_;
  const int tid  = threadIdx.x;
  const int wid  = tid >> 5;
  const int lane = tid & 31;

  const int cls = ws_cls[row];
  const v4f* __restrict__ mrow = (const v4f*)(mask_logits + (size_t)row * N_);
  const v4i* __restrict__ srow = (const v4i*)(seg_pred + (size_t)b * N_);
  v4f* __restrict__ orow = (v4f*)(masks_out + (size_t)row * N_);

  const int start4 = part * PART4;
  const int end4   = start4 + PART4;
  const int i0     = start4 + tid;

  const unsigned lds_base   = lds_off_of(&stage[0][wid][lane]);
  const unsigned lds_stride = (unsigned)sizeof(stage[0]);  // 4096 B

  // Prime the async pipeline (clamped tail addresses stay in-bounds).
#pragma unroll
  for (int j = 0; j < DEPTH; ++j) {
    int idx = i0 + j * TPB;
    idx = idx < end4 ? idx : (end4 - 1);
    async_b128(lds_base + (unsigned)(j & (DEPTH - 1)) * lds_stride,
               (const void*)(mrow + idx));
    __builtin_prefetch((const void*)(srow + idx));  // global_prefetch_b8
  }

  int cnt = 0;
  float sum = 0.0f;

  auto consume = [&](int k) {
    v4f d = stage[k & (DEPTH - 1)][wid][lane];
    int cur = i0 + k * TPB;
    if (cur < end4) {
      v4i sg = srow[cur];
      // sigmoid(x) > 0.5  <=>  x > 0
      bool s0 = (d.x > 0.0f) & (sg.x == cls);
      bool s1 = (d.y > 0.0f) & (sg.y == cls);
      bool s2 = (d.z > 0.0f) & (sg.z == cls);
      bool s3 = (d.w > 0.0f) & (sg.w == cls);
      float m0 = __fdividef(1.0f, 1.0f + __expf(-d.x));
      float m1 = __fdividef(1.0f, 1.0f + __expf(-d.y));
      float m2 = __fdividef(1.0f, 1.0f + __expf(-d.z));
      float m3 = __fdividef(1.0f, 1.0f + __expf(-d.w));
      v4f o;
      o.x = s0 ? 1.0f : 0.0f;
      o.y = s1 ? 1.0f : 0.0f;
      o.z = s2 ? 1.0f : 0.0f;
      o.w = s3 ? 1.0f : 0.0f;
      __builtin_nontemporal_store(o, orow + cur);  // streaming store
      cnt += (int)s0 + (int)s1 + (int)s2 + (int)s3;
      sum += (s0 ? m0 : 0.0f) + (s1 ? m1 : 0.0f) +
             (s2 ? m2 : 0.0f) + (s3 ? m3 : 0.0f);
    }
  };

  // Steady state: keep DEPTH loads in flight; async loads complete in order,
  // so asynccnt<=DEPTH-1 guarantees the oldest buffer is resident in LDS.
  for (int k = 0; k <= NITER - DEPTH; ++k) {
    int j = k + DEPTH;
    if (j < NITER) {
      int idx = i0 + j * TPB;
      idx = idx < end4 ? idx : (end4 - 1);
      async_b128(lds_base + (unsigned)(j & (DEPTH - 1)) * lds_stride,
                 (const void*)(mrow + idx));
      __builtin_prefetch((const void*)(srow + idx));
    }
    wait_async<DEPTH - 1>();
    consume(k);
  }
  // Drain: outstanding after consume(k) = NITER-1-k, so wait W = NITER-1-k.
#define GF_EPI(W) do { wait_async<(W)>(); consume(NITER - 1 - (W)); } while (0)
  GF_EPI(6); GF_EPI(5); GF_EPI(4); GF_EPI(3); GF_EPI(2); GF_EPI(1); GF_EPI(0);
#undef GF_EPI

  // Deterministic reduction: wave32 shuffle tree, then LDS across waves.
#pragma unroll
  for (int off = 16; off > 0; off >>= 1) {
    cnt += __shfl_xor(cnt, off, 32);
    sum += __shfl_xor(sum, off, 32);
  }
  if (lane == 0) { s_cnt[wid] = cnt; s_sum[wid] = sum; }
  __syncthreads();
  if (tid == 0) {
    int c = 0; float s = 0.0f;
#pragma unroll
    for (int i = 0; i < NWAVES; ++i) { c += s_cnt[i]; s += s_sum[i]; }
    ws_pcnt[blockIdx.x] = c;   // blockIdx.x == row*SPLIT + part
    ws_psum[blockIdx.x] = s;
  }
}

// ---- Kernel C: fold partials -> scores / valid --------------------------
__global__ __launch_bounds__(TPB) void gf_finalize(
    const int* __restrict__ ws_pcnt, const float* __restrict__ ws_psum,
    const int* __restrict__ ws_cls, float* __restrict__ scores_out,
    float* __restrict__ valid_out, int* __restrict__ ws_valid) {
  int row = blockIdx.x * TPB + threadIdx.x;
  if (row >= ROWS) return;
  int c = 0; float s = 0.0f;
#pragma unroll
  for (int p = 0; p < SPLIT; ++p) {
    c += ws_pcnt[row * SPLIT + p];
    s += ws_psum[row * SPLIT + p];
  }
  int cls = ws_cls[row];
  int valid = (c >= THRESH_CNT) && (cls >= MIN_CLS);
  scores_out[row] = valid ? (s / (float)(c > 1 ? c : 1)) : 0.0f;
  valid_out[row] = valid ? 1.0f : 0.0f;
  ws_valid[row] = valid;
}

// ---- Kernel D: zero mask rows whose query is invalid --------------------
__global__ __launch_bounds__(TPB) void gf_fixup(
    const int* __restrict__ ws_valid, float* __restrict__ masks_out) {
  int row  = blockIdx.x / SPLIT;
  int part = blockIdx.x % SPLIT;
  __shared__ int sv;
  if (threadIdx.x == 0) sv = ws_valid[row];
  __syncthreads();
  if (sv) return;  // ~80% of blocks exit here
  v4f* __restrict__ orow = (v4f*)(masks_out + (size_t)row * N_);
  int i0 = part * PART4 + threadIdx.x;
  int end4 = part * PART4 + PART4;
  v4f z = {0.0f, 0.0f, 0.0f, 0.0f};
  for (int k = 0; k < NITER; ++k) {
    int cur = i0 + k * TPB;
    if (cur < end4) __builtin_nontemporal_store(z, orow + cur);
  }
}

extern "C" void kernel_launch(void* const* d_in, const int* in_sizes, int n_in,
                              void* d_out, int out_size, void* d_ws,
                              size_t ws_size, hipStream_t stream) {
  const float* mask_logits = (const float*)d_in[0];  // [B,Q,N]
  const float* cls_logits  = (const float*)d_in[1];  // [B,Q,C]
  const int*   seg_pred    = (const int*)d_in[2];    // [B,N]
  const int*   fg_idxs     = (const int*)d_in[3];    // [B*N]

  // Outputs concatenated flat in return order, all float:
  float* out        = (float*)d_out;
  float* masks_out  = out;                                   // B*Q*N
  float* scores_out = out + (size_t)ROWS * N_;               // 512
  float* valid_out  = scores_out + ROWS;                     // 512
  float* cls_out    = valid_out + ROWS;                      // 512
  float* gids_out   = cls_out + ROWS;                        // B*N

  char* ws = (char*)d_ws;                   // ~22.5 KB used, fully rewritten
  int*   ws_cls   = (int*)(ws);             // 512
  int*   ws_pcnt  = (int*)(ws + 4096);      // 2048
  float* ws_psum  = (float*)(ws + 12288);   // 2048
  int*   ws_valid = (int*)(ws + 20480);     // 512

  gf_prep<<<(B_ * N_ + TPB - 1) / TPB, TPB, 0, stream>>>(
      cls_logits, fg_idxs, ws_cls, cls_out, gids_out);
  gf_main<<<ROWS * SPLIT, TPB, 0, stream>>>(
      mask_logits, seg_pred, ws_cls, masks_out, ws_pcnt, ws_psum);
  gf_finalize<<<(ROWS + TPB - 1) / TPB, TPB, 0, stream>>>(
      ws_pcnt, ws_psum, ws_cls, scores_out, valid_out, ws_valid);
  gf_fixup<<<ROWS * SPLIT, TPB, 0, stream>>>(ws_valid, masks_out);
}